// kNNQuery_43997644980253
// MI455X (gfx1250) — compile-verified
//
#include <hip/hip_runtime.h>

#define B_    4
#define C_    64
#define NQ_   8192
#define NK_   8192
#define KNN   16
#define CHUNK 2048
#define QPB   128   // queries per block = 8 waves * 16

typedef __attribute__((ext_vector_type(2))) float v2f;
typedef __attribute__((ext_vector_type(8))) float v8f;

// Branchless bitonic compare-exchange across 16-lane groups (wave32 halves independent).
// Sort key = (dist, idx) lexicographic ascending. Lowers to bpermute + cmp + cndmask.
__device__ __forceinline__ void cmpx(float& d, int& i, int j16, int mask, int k) {
    float od = __shfl_xor(d, mask, 16);
    int   oi = __shfl_xor(i, mask, 16);
    bool lower = ((j16 & mask) == 0);
    bool up    = ((j16 & k) == 0);          // k==16 -> always ascending
    bool oless = (od < d) | ((od == d) & (oi < i));
    bool take  = (oless == (lower == up));
    d = take ? od : d;
    i = take ? oi : i;
}

__global__ __launch_bounds__(256)
void knn_gather_kernel(const float* __restrict__ qc, const float* __restrict__ qfeat,
                       const float* __restrict__ kc, const float* __restrict__ kfeat,
                       float* __restrict__ out) {
    __shared__ float skx[CHUNK], sky[CHUNK], skz[CHUNK], skq[CHUNK];
    __shared__ int   sidx[QPB * KNN];

    const int b    = blockIdx.y;
    const int q0   = blockIdx.x * QPB;
    const int t    = threadIdx.x;
    const int wv   = t >> 5;
    const int lane = t & 31;
    const int j16  = lane & 15;
    const int h    = lane >> 4;

    // ---- A matrix: this wave's 16 queries, coords scaled by -2, K padded 3->4.
    // 16x4 f32 A layout: lanes 0-15 hold K=0 (v0) / K=1 (v1); lanes 16-31 hold K=2 / K=3.
    const int qw = q0 + wv * 16;
    const size_t qcb = (size_t)b * 3 * NQ_;
    float qx = qc[qcb + 0 * NQ_ + qw + j16];
    float qy = qc[qcb + 1 * NQ_ + qw + j16];
    float qz = qc[qcb + 2 * NQ_ + qw + j16];
    v2f Av;
    Av[0] = (h == 0) ? (-2.0f * qx) : (-2.0f * qz);
    Av[1] = (h == 0) ? (-2.0f * qy) : 0.0f;

    // Per-query top-16 list, striped across the 16 lanes of each half:
    // row r (query m = wv*16 + r + 8*h), entry j16 = rank. Sorted ascending.
    float Ld[8]; int Li[8]; float worst[8];
#pragma unroll
    for (int r = 0; r < 8; ++r) { Ld[r] = 3.0e38f; Li[r] = 0; worst[r] = 3.0e38f; }

    const size_t kcb = (size_t)b * 3 * NK_;
    for (int ch = 0; ch < NK_; ch += CHUNK) {
        __syncthreads();
        for (int s = t; s < CHUNK; s += 256) {
            float x = kc[kcb + 0 * NK_ + ch + s];
            float y = kc[kcb + 1 * NK_ + ch + s];
            float z = kc[kcb + 2 * NK_ + ch + s];
            skx[s] = x; sky[s] = y; skz[s] = z;
            skq[s] = x * x + y * y + z * z;
        }
        // prefetch next chunk of key coords into L2 (global_prefetch_b8)
        if (ch + CHUNK < NK_) {
            const float* nxt = kc + kcb + ch + CHUNK;
            __builtin_prefetch(nxt + 0 * NK_ + t * (CHUNK / 256), 0, 1);
            __builtin_prefetch(nxt + 1 * NK_ + t * (CHUNK / 256), 0, 1);
            __builtin_prefetch(nxt + 2 * NK_ + t * (CHUNK / 256), 0, 1);
        }
        __syncthreads();

#pragma unroll 2
        for (int tb = 0; tb < CHUNK; tb += 16) {
            const int kk = tb + j16;
            float kx = skx[kk], ky = sky[kk], kz = skz[kk], kq = skq[kk];
            // B 4x16 layout: VGPR v, lanes 0-15 -> K=v row, lanes 16-31 -> K=v+2 row.
            v2f Bv;
            Bv[0] = (h == 0) ? kx : kz;
            Bv[1] = (h == 0) ? ky : 0.0f;
            v8f acc = {};
            acc = __builtin_amdgcn_wmma_f32_16x16x4_f32(
                false, Av, false, Bv, (short)0, acc, false, false);

            const int kidx = ch + tb + j16;
            float cd[8];
            unsigned long long mk[8];
            unsigned long long mAny = 0ull;
#pragma unroll
            for (int r = 0; r < 8; ++r) {
                cd[r] = acc[r] + kq;             // -2*q.k + |k|^2 (|q|^2 dropped: order-invariant)
                mk[r] = __ballot(cd[r] < worst[r]);  // v_cmp -> sgpr mask
                mAny |= mk[r];                       // scalar OR reduction
            }
            if (mAny) {                           // rare after threshold tightens
#pragma unroll
                for (int r = 0; r < 8; ++r) {
                    if (mk[r]) {
                        // full bitonic sort of 16 candidates, ascending
                        float sd = cd[r]; int si = kidx;
                        cmpx(sd, si, j16, 1, 2);
                        cmpx(sd, si, j16, 2, 4);  cmpx(sd, si, j16, 1, 4);
                        cmpx(sd, si, j16, 4, 8);  cmpx(sd, si, j16, 2, 8);  cmpx(sd, si, j16, 1, 8);
                        cmpx(sd, si, j16, 8, 16); cmpx(sd, si, j16, 4, 16);
                        cmpx(sd, si, j16, 2, 16); cmpx(sd, si, j16, 1, 16);
                        // pair list[j] with reversed candidate[15-j], keep min -> bitonic seq
                        float rd = __shfl_xor(sd, 15, 16);
                        int   ri = __shfl_xor(si, 15, 16);
                        bool rl = (rd < Ld[r]) | ((rd == Ld[r]) & (ri < Li[r]));
                        Ld[r] = rl ? rd : Ld[r];
                        Li[r] = rl ? ri : Li[r];
                        // 4-stage bitonic merge back to sorted ascending
                        cmpx(Ld[r], Li[r], j16, 8, 16);
                        cmpx(Ld[r], Li[r], j16, 4, 16);
                        cmpx(Ld[r], Li[r], j16, 2, 16);
                        cmpx(Ld[r], Li[r], j16, 1, 16);
                        worst[r] = __shfl(Ld[r], 15, 16);
                    }
                }
            }
        }
    }

    // ---- publish indices to LDS for the gather phase
    __syncthreads();
#pragma unroll
    for (int r = 0; r < 8; ++r) {
        int ql = wv * 16 + r + 8 * h;
        sidx[ql * KNN + j16] = Li[r];
    }
    __syncthreads();

    // ---- gather + subtract + write (bandwidth phase)
    // thread -> (query_local, half of K); 16 contiguous floats per (c,q) via float4s.
    const int ql = t >> 1;
    const int jh = (t & 1) * 8;
    const int q  = q0 + ql;
    const size_t fb = (size_t)b * C_;
    for (int c = 0; c < C_; ++c) {
        float qv = qfeat[(fb + c) * NQ_ + q];
        const float* kfr = kfeat + (fb + c) * NK_;   // 32KB row: L2-resident gathers
        float rr[8];
#pragma unroll
        for (int jj = 0; jj < 8; ++jj) {
            int ki = sidx[ql * KNN + jh + jj];
            rr[jj] = kfr[ki] - qv;
        }
        size_t o0 = (((size_t)(b * 2 * C_ + c)       * NQ_) + q) * KNN + jh;
        size_t o1 = (((size_t)(b * 2 * C_ + C_ + c)  * NQ_) + q) * KNN + jh;
        float4* p0 = (float4*)(out + o0);
        p0[0] = make_float4(rr[0], rr[1], rr[2], rr[3]);
        p0[1] = make_float4(rr[4], rr[5], rr[6], rr[7]);
        float4 qv4 = make_float4(qv, qv, qv, qv);
        float4* p1 = (float4*)(out + o1);
        p1[0] = qv4;
        p1[1] = qv4;
    }
}

extern "C" void kernel_launch(void* const* d_in, const int* in_sizes, int n_in,
                              void* d_out, int out_size, void* d_ws, size_t ws_size,
                              hipStream_t stream) {
    (void)in_sizes; (void)n_in; (void)d_ws; (void)ws_size; (void)out_size;
    const float* qc = (const float*)d_in[0];
    const float* qf = (const float*)d_in[1];
    const float* kc = (const float*)d_in[2];
    const float* kf = (const float*)d_in[3];
    float* out = (float*)d_out;
    dim3 grid(NQ_ / QPB, B_);
    knn_gather_kernel<<<grid, dim3(256), 0, stream>>>(qc, qf, kc, kf, out);
}